// HypergraphNet_1271310320103
// MI455X (gfx1250) — compile-verified
//
#include <hip/hip_runtime.h>
#include <cstdint>
#include <cstddef>

// Problem constants (fixed by the reference)
#define N_NODES 50000
#define N_EDGES 10000
#define NNZV    250000
#define DMODEL  256      // heads * out_ch, identical for all three layers
#define HEADSN  4
#define CHEAD   64

typedef __attribute__((ext_vector_type(2))) float v2f;
typedef __attribute__((ext_vector_type(8))) float v8f;

// ------------------------------------------------------------------
// GEMM: C[M,Nout] = A[M,K] @ B[K,Nout] (+ bias), fp32 via WMMA 16x16x4.
// One wave computes a 16x64 tile (4 accumulators); A fragment reused 4x.
// A layout (ISA 7.12.2, 32-bit A 16x4): lanes 0-15 hold row=lane, K={k0,k0+1};
// lanes 16-31 hold row=lane-16, K={k0+2,k0+3}. C layout per spec.
// ------------------------------------------------------------------
__global__ void k_gemm_wmma(const float* __restrict__ A,
                            const float* __restrict__ Bw,
                            const float* __restrict__ bias,
                            float* __restrict__ C,
                            int M, int K, int Nout)
{
  const int lane = threadIdx.x & 31;
  const int wid  = (int)((blockIdx.x * blockDim.x + threadIdx.x) >> 5);
  const int nb   = Nout >> 6;                 // 64-wide N blocks
  const int mt   = wid / nb;
  const int nt   = wid - mt * nb;
  if (mt >= (M >> 4)) return;                 // wave-uniform exit: EXEC stays full
  const int m0 = mt << 4;
  const int n0 = nt << 6;

  const int hi  = lane >> 4;                  // half-wave select
  const int l15 = lane & 15;
  const int kb  = hi << 1;                    // K sub-offset 0 or 2

  v8f acc0 = {}, acc1 = {}, acc2 = {}, acc3 = {};

  const float* ap  = A  + (size_t)(m0 + l15) * K + kb;
  const float* bp0 = Bw + (size_t)kb * Nout + n0 + l15;

  for (int k0 = 0; k0 < K; k0 += 4) {
    v2f a; a.x = ap[k0]; a.y = ap[k0 + 1];
    const float* bp = bp0 + (size_t)k0 * Nout;
    v2f b0; b0.x = bp[0];   b0.y = bp[Nout];
    v2f b1; b1.x = bp[16];  b1.y = bp[Nout + 16];
    v2f b2; b2.x = bp[32];  b2.y = bp[Nout + 32];
    v2f b3; b3.x = bp[48];  b3.y = bp[Nout + 48];
    acc0 = __builtin_amdgcn_wmma_f32_16x16x4_f32(false, a, false, b0, (short)0, acc0, false, false);
    acc1 = __builtin_amdgcn_wmma_f32_16x16x4_f32(false, a, false, b1, (short)0, acc1, false, false);
    acc2 = __builtin_amdgcn_wmma_f32_16x16x4_f32(false, a, false, b2, (short)0, acc2, false, false);
    acc3 = __builtin_amdgcn_wmma_f32_16x16x4_f32(false, a, false, b3, (short)0, acc3, false, false);
  }

  float bv0 = 0.f, bv1 = 0.f, bv2 = 0.f, bv3 = 0.f;
  if (bias) {
    bv0 = bias[n0 + l15];       bv1 = bias[n0 + 16 + l15];
    bv2 = bias[n0 + 32 + l15];  bv3 = bias[n0 + 48 + l15];
  }
  const int r0 = m0 + (hi << 3);
  #pragma unroll
  for (int v = 0; v < 8; ++v) {
    float* cp = C + (size_t)(r0 + v) * Nout + n0 + l15;
    cp[0]  = acc0[v] + bv0;
    cp[16] = acc1[v] + bv1;
    cp[32] = acc2[v] + bv2;
    cp[48] = acc3[v] + bv3;
  }
}

// ---------------- fill / invert / bias-init ------------------------
__global__ void k_fill(float* __restrict__ p, float v, int n) {
  int i = blockIdx.x * blockDim.x + threadIdx.x;
  if (i < n) p[i] = v;
}
__global__ void k_invert(float* __restrict__ p, int n) {
  int i = blockIdx.x * blockDim.x + threadIdx.x;
  if (i < n) { float v = p[i]; p[i] = (v > 0.f) ? (1.f / v) : 0.f; }
}
__global__ void k_init_bias(float* __restrict__ out, const float* __restrict__ bias, int n) {
  int i = blockIdx.x * blockDim.x + threadIdx.x;
  if (i < n) out[i] = bias[i & (DMODEL - 1)];
}

// ---------------- degrees: D (per node), B (per edge) --------------
__global__ void k_degrees(const long long* __restrict__ nidx,
                          const long long* __restrict__ eidx,
                          const float* __restrict__ hew,
                          float* __restrict__ D, float* __restrict__ B, int nnz) {
  int i = blockIdx.x * blockDim.x + threadIdx.x;
  if (i >= nnz) return;
  long long n = nidx[i], e = eidx[i];
  atomicAdd(&D[n], hew[e]);
  atomicAdd(&B[e], 1.0f);
}

// ---------------- per-row attention dots ---------------------------
// S[r*4+h] = dot(X[r, h*64 : h*64+64], att[h, part*64 : part*64+64])
__global__ void k_rowdot(const float* __restrict__ X, const float* __restrict__ att,
                         int part, float* __restrict__ S, int rows) {
  int t = blockIdx.x * blockDim.x + threadIdx.x;
  if (t >= rows * HEADSN) return;
  int r = t >> 2, h = t & 3;
  const float4* xp = (const float4*)(X + (size_t)r * DMODEL + h * CHEAD);
  const float4* apv = (const float4*)(att + h * (2 * CHEAD) + part * CHEAD);
  float s = 0.f;
  #pragma unroll
  for (int j = 0; j < CHEAD / 4; ++j) {
    float4 xv = xp[j], av = apv[j];
    s += xv.x * av.x + xv.y * av.y + xv.z * av.z + xv.w * av.w;
  }
  S[t] = s;
}

// ---------------- alpha = relu(sx+se); segment max -----------------
// alpha >= 0 after relu, so float bits compare monotonically as u32
// and the 0-init matches reference's where(isfinite, amax, 0).
__global__ void k_alpha_max(const long long* __restrict__ nidx,
                            const long long* __restrict__ eidx,
                            const float* __restrict__ sx, const float* __restrict__ se,
                            float* __restrict__ alpha, unsigned* __restrict__ amax, int nnz) {
  int i = blockIdx.x * blockDim.x + threadIdx.x;
  if (i >= nnz) return;
  long long n = nidx[i], e = eidx[i];
  #pragma unroll
  for (int h = 0; h < HEADSN; ++h) {
    float a = sx[n * 4 + h] + se[e * 4 + h];
    a = a > 0.f ? a : 0.f;                    // leaky_relu with slope 0
    alpha[(size_t)i * 4 + h] = a;
    atomicMax(&amax[n * 4 + h], __float_as_uint(a));
  }
}

// ---------------- exp + segment sum --------------------------------
__global__ void k_alpha_exp(const long long* __restrict__ nidx,
                            float* __restrict__ alpha, const unsigned* __restrict__ amax,
                            float* __restrict__ ssum, int nnz) {
  int i = blockIdx.x * blockDim.x + threadIdx.x;
  if (i >= nnz) return;
  long long n = nidx[i];
  #pragma unroll
  for (int h = 0; h < HEADSN; ++h) {
    float m  = __uint_as_float(amax[n * 4 + h]);
    float ex = __expf(alpha[(size_t)i * 4 + h] - m);
    alpha[(size_t)i * 4 + h] = ex;            // keep un-normalized exp; normalize at use
    atomicAdd(&ssum[n * 4 + h], ex);
  }
}

// ---------------- aggregation: node -> edge ------------------------
// one wave per incidence; lanes cover 256 channels in 8 strided steps
__global__ void k_agg_edge(const long long* __restrict__ nidx,
                           const long long* __restrict__ eidx,
                           const float* __restrict__ xl, const float* __restrict__ alpha,
                           const float* __restrict__ ssum, const float* __restrict__ Binv,
                           float* __restrict__ out_e, int nnz) {
  int w = (int)((blockIdx.x * blockDim.x + threadIdx.x) >> 5);
  if (w >= nnz) return;
  int lane = threadIdx.x & 31;
  long long n = nidx[w], e = eidx[w];
  float bi = Binv[e];
  #pragma unroll
  for (int it = 0; it < 8; ++it) {
    int ch = it * 32 + lane;
    int h  = ch >> 6;
    float a = alpha[(size_t)w * 4 + h] / (ssum[n * 4 + h] + 1e-16f);
    float v = bi * a * xl[(size_t)n * DMODEL + ch];
    atomicAdd(&out_e[(size_t)e * DMODEL + ch], v);
  }
}

// ---------------- aggregation: edge -> node ------------------------
__global__ void k_agg_node(const long long* __restrict__ nidx,
                           const long long* __restrict__ eidx,
                           const float* __restrict__ out_e, const float* __restrict__ alpha,
                           const float* __restrict__ ssum, const float* __restrict__ Dinv,
                           float* __restrict__ out, int nnz) {
  int w = (int)((blockIdx.x * blockDim.x + threadIdx.x) >> 5);
  if (w >= nnz) return;
  int lane = threadIdx.x & 31;
  long long n = nidx[w], e = eidx[w];
  float di = Dinv[n];
  #pragma unroll
  for (int it = 0; it < 8; ++it) {
    int ch = it * 32 + lane;
    int h  = ch >> 6;
    float a = alpha[(size_t)w * 4 + h] / (ssum[n * 4 + h] + 1e-16f);
    float v = di * a * out_e[(size_t)e * DMODEL + ch];
    atomicAdd(&out[(size_t)n * DMODEL + ch], v);
  }
}

// ==================================================================
// Host side
// ==================================================================
static inline int cdiv(int a, int b) { return (a + b - 1) / b; }

static void gemm(hipStream_t s, const float* A, const float* B, const float* bias,
                 float* C, int M, int K, int Nout) {
  int waves  = (M >> 4) * (Nout >> 6);
  int blocks = cdiv(waves * 32, 256);
  k_gemm_wmma<<<blocks, 256, 0, s>>>(A, B, bias, C, M, K, Nout);
}

static void hconv(hipStream_t s, const long long* nidx, const long long* eidx,
                  const float* xl, const float* el, const float* att, const float* bias,
                  const float* Dinv, const float* Binv,
                  float* sx, float* se, float* alpha, unsigned* amax, float* ssum,
                  float* out_e, float* out) {
  k_rowdot<<<cdiv(N_NODES * HEADSN, 256), 256, 0, s>>>(xl, att, 0, sx, N_NODES);
  k_rowdot<<<cdiv(N_EDGES * HEADSN, 256), 256, 0, s>>>(el, att, 1, se, N_EDGES);
  k_fill<<<cdiv(N_NODES * HEADSN, 256), 256, 0, s>>>((float*)amax, 0.f, N_NODES * HEADSN);
  k_fill<<<cdiv(N_NODES * HEADSN, 256), 256, 0, s>>>(ssum, 0.f, N_NODES * HEADSN);
  k_fill<<<cdiv(N_EDGES * DMODEL, 256), 256, 0, s>>>(out_e, 0.f, N_EDGES * DMODEL);
  k_alpha_max<<<cdiv(NNZV, 256), 256, 0, s>>>(nidx, eidx, sx, se, alpha, amax, NNZV);
  k_alpha_exp<<<cdiv(NNZV, 256), 256, 0, s>>>(nidx, alpha, amax, ssum, NNZV);
  k_init_bias<<<cdiv(N_NODES * DMODEL, 256), 256, 0, s>>>(out, bias, N_NODES * DMODEL);
  k_agg_edge<<<cdiv(NNZV * 32, 256), 256, 0, s>>>(nidx, eidx, xl, alpha, ssum, Binv, out_e, NNZV);
  k_agg_node<<<cdiv(NNZV * 32, 256), 256, 0, s>>>(nidx, eidx, out_e, alpha, ssum, Dinv, out, NNZV);
}

extern "C" void kernel_launch(void* const* d_in, const int* in_sizes, int n_in,
                              void* d_out, int out_size, void* d_ws, size_t ws_size,
                              hipStream_t stream) {
  (void)in_sizes; (void)n_in; (void)out_size; (void)ws_size;

  const float*     x     = (const float*)d_in[0];
  const long long* hei   = (const long long*)d_in[1];   // int64 (2, NNZ)
  const float*     hew   = (const float*)d_in[2];
  const float*     hef   = (const float*)d_in[3];
  const float*     lin1W = (const float*)d_in[4];
  const float*     lin1b = (const float*)d_in[5];
  const float*     lin2W = (const float*)d_in[6];
  const float*     lin2b = (const float*)d_in[7];
  const float*     lin3W = (const float*)d_in[8];
  const float*     lin3b = (const float*)d_in[9];
  const float*     W1    = (const float*)d_in[10];
  const float*     att1  = (const float*)d_in[11];
  const float*     b1    = (const float*)d_in[12];
  const float*     W2    = (const float*)d_in[13];
  const float*     att2  = (const float*)d_in[14];
  const float*     b2    = (const float*)d_in[15];
  const float*     W3    = (const float*)d_in[16];
  const float*     att3  = (const float*)d_in[17];
  const float*     b3    = (const float*)d_in[18];

  const long long* nidx = hei;            // row 0: node indices
  const long long* eidx = hei + NNZV;     // row 1: edge indices

  // -------- workspace carve-up (256B aligned slices) --------
  uint8_t* base = (uint8_t*)d_ws;
  size_t off = 0;
  auto take = [&](size_t bytes) -> void* {
    void* p = base + off;
    off += (bytes + 255) & ~(size_t)255;
    return p;
  };
  float*    heA   = (float*)take((size_t)N_EDGES * DMODEL * 4);
  float*    heB   = (float*)take((size_t)N_EDGES * DMODEL * 4);
  float*    el    = (float*)take((size_t)N_EDGES * DMODEL * 4);
  float*    out_e = (float*)take((size_t)N_EDGES * DMODEL * 4);
  float*    xl    = (float*)take((size_t)N_NODES * DMODEL * 4);
  float*    xcur  = (float*)take((size_t)N_NODES * DMODEL * 4);
  float*    sx    = (float*)take((size_t)N_NODES * HEADSN * 4);
  float*    se    = (float*)take((size_t)N_EDGES * HEADSN * 4);
  float*    alpha = (float*)take((size_t)NNZV * HEADSN * 4);
  unsigned* amax  = (unsigned*)take((size_t)N_NODES * HEADSN * 4);
  float*    ssum  = (float*)take((size_t)N_NODES * HEADSN * 4);
  float*    Dinv  = (float*)take((size_t)N_NODES * 4);
  float*    Binv  = (float*)take((size_t)N_EDGES * 4);

  // -------- degree normalizations (layer-invariant: compute once) --------
  k_fill<<<cdiv(N_NODES, 256), 256, 0, stream>>>(Dinv, 0.f, N_NODES);
  k_fill<<<cdiv(N_EDGES, 256), 256, 0, stream>>>(Binv, 0.f, N_EDGES);
  k_degrees<<<cdiv(NNZV, 256), 256, 0, stream>>>(nidx, eidx, hew, Dinv, Binv, NNZV);
  k_invert<<<cdiv(N_NODES, 256), 256, 0, stream>>>(Dinv, N_NODES);
  k_invert<<<cdiv(N_EDGES, 256), 256, 0, stream>>>(Binv, N_EDGES);

  // -------- layer 1 --------
  gemm(stream, hef, lin1W, lin1b, heA, N_EDGES, 128, 128);   // he1 = hef@lin1_W+b   (E,128)
  gemm(stream, heA, W1, nullptr, el, N_EDGES, 128, 256);     // el1 = he1@W1         (E,256)
  gemm(stream, heA, lin2W, lin2b, heB, N_EDGES, 128, 256);   // he2 = he1@lin2_W+b   (E,256)
  gemm(stream, x, W1, nullptr, xl, N_NODES, 128, 256);       // xl1 = x@W1           (N,256)
  hconv(stream, nidx, eidx, xl, el, att1, b1, Dinv, Binv,
        sx, se, alpha, amax, ssum, out_e, xcur);             // x2 -> xcur

  // -------- layer 2 --------
  gemm(stream, heB, W2, nullptr, el, N_EDGES, 256, 256);     // el2 = he2@W2
  gemm(stream, heB, lin3W, lin3b, heA, N_EDGES, 256, 256);   // he3 = he2@lin3_W+b
  gemm(stream, xcur, W2, nullptr, xl, N_NODES, 256, 256);    // xl2 = x2@W2
  hconv(stream, nidx, eidx, xl, el, att2, b2, Dinv, Binv,
        sx, se, alpha, amax, ssum, out_e, xcur);             // x3 -> xcur (xl already consumed it)

  // -------- layer 3 --------
  gemm(stream, heA, W3, nullptr, el, N_EDGES, 256, 256);     // el3 = he3@W3
  gemm(stream, xcur, W3, nullptr, xl, N_NODES, 256, 256);    // xl3 = x3@W3
  hconv(stream, nidx, eidx, xl, el, att3, b3, Dinv, Binv,
        sx, se, alpha, amax, ssum, out_e, (float*)d_out);    // final output
}